// RuchbahMoEGate_4131758538902
// MI455X (gfx1250) — compile-verified
//
#include <hip/hip_runtime.h>

typedef __attribute__((ext_vector_type(2))) float v2f;
typedef __attribute__((ext_vector_type(8))) float v8f;

#define HIDDEN      2048
#define NEXP        64
#define TOK_PER_BLK 128
#define LDS_STRIDE  68   // 64 experts + pad; 68 = 4 mod 64 -> conflict-free column sums
#define WS_STRIDE   72   // per-block: 64 expert-load partials + 1 z-loss partial + pad

// ---------------------------------------------------------------------------
// Main kernel: WMMA f32 GEMM (logits = x * W^T), softmax, top-2, loss partials
// grid = ntok/128 blocks, block = 256 threads (8 waves, 16 tokens per wave)
// ---------------------------------------------------------------------------
__global__ __launch_bounds__(256) void moe_gate_main(
    const float* __restrict__ x, const float* __restrict__ W,
    float* __restrict__ out_scores, int* __restrict__ out_idx,
    float* __restrict__ ws_partials, int ntok)
{
    __shared__ float lds[TOK_PER_BLK * LDS_STRIDE];
    __shared__ float zbuf[TOK_PER_BLK];

    const int tid  = threadIdx.x;
    const int wave = tid >> 5;
    const int lane = tid & 31;
    const int hf   = lane >> 4;   // half-wave select (K phase)
    const int lm   = lane & 15;   // token (A) / expert (B) within tile

    // ---------------- Phase 1: GEMM via V_WMMA_F32_16X16X4_F32 -------------
    // A 16x4 layout: lane l holds x[m = l%16][k + 2*hf + {0,1}] in 2 VGPRs
    // B 4x16 layout: lane l holds W[n = l%16][k + 2*hf + {0,1}] in 2 VGPRs
    const int tokBase = blockIdx.x * TOK_PER_BLK + wave * 16;
    const float* xa = x + (size_t)(tokBase + lm) * HIDDEN + 2 * hf;
    const float* wb = W + (size_t)lm * HIDDEN + 2 * hf;

    v8f c0 = {}, c1 = {}, c2 = {}, c3 = {};
#pragma unroll 4
    for (int k = 0; k < HIDDEN; k += 4) {
        v2f a  = *(const v2f*)(xa + k);
        v2f b0 = *(const v2f*)(wb + k);
        v2f b1 = *(const v2f*)(wb + 16 * HIDDEN + k);
        v2f b2 = *(const v2f*)(wb + 32 * HIDDEN + k);
        v2f b3 = *(const v2f*)(wb + 48 * HIDDEN + k);
        c0 = __builtin_amdgcn_wmma_f32_16x16x4_f32(false, a, false, b0, (short)0, c0, false, false);
        c1 = __builtin_amdgcn_wmma_f32_16x16x4_f32(false, a, false, b1, (short)0, c1, false, false);
        c2 = __builtin_amdgcn_wmma_f32_16x16x4_f32(false, a, false, b2, (short)0, c2, false, false);
        c3 = __builtin_amdgcn_wmma_f32_16x16x4_f32(false, a, false, b3, (short)0, c3, false, false);
    }

    // C/D layout: VGPR j, lane l -> token (wave*16 + j + 8*hf), expert (tile*16 + l%16)
    const int mb = wave * 16 + 8 * hf;
#pragma unroll
    for (int j = 0; j < 8; ++j) {
        lds[(mb + j) * LDS_STRIDE +  0 + lm] = c0[j];
        lds[(mb + j) * LDS_STRIDE + 16 + lm] = c1[j];
        lds[(mb + j) * LDS_STRIDE + 32 + lm] = c2[j];
        lds[(mb + j) * LDS_STRIDE + 48 + lm] = c3[j];
    }
    __syncthreads();

    // ---------------- Phase 2: per-token softmax / top-2 / z-loss ----------
    if (tid < TOK_PER_BLK) {
        float l[NEXP];
        const float4* row = (const float4*)(&lds[tid * LDS_STRIDE]); // 272B row stride -> 16B aligned
#pragma unroll
        for (int i = 0; i < NEXP / 4; ++i) {
            float4 v = row[i];
            l[4*i+0] = v.x; l[4*i+1] = v.y; l[4*i+2] = v.z; l[4*i+3] = v.w;
        }
        // temperature == 1.0 (clamped buffer init) -> no scaling
        float m = l[0];
#pragma unroll
        for (int i = 1; i < NEXP; ++i) m = fmaxf(m, l[i]);
        float s = 0.0f;
#pragma unroll
        for (int i = 0; i < NEXP; ++i) { float e = __expf(l[i] - m); l[i] = e; s += e; }
        const float inv_s = 1.0f / s;

        // top-2 (monotonic under exp; strict > keeps lowest index on ties, like lax.top_k)
        float v1 = l[0], v2 = -1.0f;
        int   i1 = 0,    i2 = 0;
#pragma unroll
        for (int i = 1; i < NEXP; ++i) {
            float li = l[i];
            if (li > v1)      { v2 = v1; i2 = i1; v1 = li; i1 = i; }
            else if (li > v2) { v2 = li; i2 = i; }
        }
        const float p1 = v1 * inv_s, p2 = v2 * inv_s;
        const float w1 = 1.0f / (1.0f + __expf(p2 - p1));  // softmax over top-2 *scores*
        const float w2 = 1.0f - w1;

        const int g = blockIdx.x * TOK_PER_BLK + tid;
        out_scores[2 * g + 0] = w1;
        out_scores[2 * g + 1] = w2;
        out_idx[2 * g + 0] = i1;
        out_idx[2 * g + 1] = i2;

        // write probabilities back for expert-load reduction
#pragma unroll
        for (int i = 0; i < NEXP; ++i) lds[tid * LDS_STRIDE + i] = l[i] * inv_s;

        const float lse = m + __logf(s);
        zbuf[tid] = lse * lse;
    }
    __syncthreads();

    // ---------------- Phase 3: deterministic per-block partial sums --------
    if (tid < NEXP) {
        float s = 0.0f;
        for (int r = 0; r < TOK_PER_BLK; ++r) s += lds[r * LDS_STRIDE + tid];
        ws_partials[blockIdx.x * WS_STRIDE + tid] = s;
    } else if (tid == NEXP) {
        float z = 0.0f;
        for (int r = 0; r < TOK_PER_BLK; ++r) z += zbuf[r];
        ws_partials[blockIdx.x * WS_STRIDE + NEXP] = z;
    }
}

// ---------------------------------------------------------------------------
// Finalize: reduce per-block partials -> scalar total loss
// ---------------------------------------------------------------------------
__global__ __launch_bounds__(64) void moe_gate_finalize(
    const float* __restrict__ ws_partials, float* __restrict__ loss_out,
    int nblk, int ntok)
{
    __shared__ float red[NEXP];
    const int e = threadIdx.x;
    float s = 0.0f;
    for (int b = 0; b < nblk; ++b) s += ws_partials[b * WS_STRIDE + e];
    const float load = s / (float)ntok;
    const float d = load - 1.0f / (float)NEXP;
    red[e] = d * d;
    __syncthreads();
    if (e == 0) {
        float lb = 0.0f;
#pragma unroll
        for (int i = 0; i < NEXP; ++i) lb += red[i];
        float z = 0.0f;
        for (int b = 0; b < nblk; ++b) z += ws_partials[b * WS_STRIDE + NEXP];
        loss_out[0] = 0.01f * lb + 1e-4f * (z / (float)ntok);
    }
}

extern "C" void kernel_launch(void* const* d_in, const int* in_sizes, int n_in,
                              void* d_out, int out_size, void* d_ws, size_t ws_size,
                              hipStream_t stream) {
    const float* x = (const float*)d_in[0];   // [4*4096, 2048] f32
    const float* W = (const float*)d_in[1];   // [64, 2048] f32
    const int ntok = in_sizes[0] / HIDDEN;    // 16384
    const int nblk = ntok / TOK_PER_BLK;      // 128

    float* out        = (float*)d_out;
    float* out_scores = out;                         // [ntok, 2] f32
    int*   out_idx    = (int*)(out + 2 * ntok);      // [ntok, 2] i32 (bit-cast slot)
    float* loss       = out + 4 * ntok;              // [1]
    float* ws         = (float*)d_ws;                // nblk * WS_STRIDE floats

    moe_gate_main<<<nblk, 256, 0, stream>>>(x, W, out_scores, out_idx, ws, ntok);
    moe_gate_finalize<<<1, 64, 0, stream>>>(ws, loss, nblk, ntok);
}